// SpectralConv3d_53824530154103
// MI455X (gfx1250) — compile-verified
//
#include <hip/hip_runtime.h>
#include <math.h>

typedef __attribute__((ext_vector_type(2))) float v2f;
typedef __attribute__((ext_vector_type(8))) float v8f;

#define PI2 6.283185307179586476925f

// ---------------------------------------------------------------------------
// Transform tables (computed on device each call; deterministic):
//  C64/S64/NS64: 64x64 cos / sin / -sin DFT matrices
//  Ci/Si/NSi:    64x8 expansion matrices (zero-padded inverse side)
//  Tr/Ti:        64x8 fused axis-3 expansion + Hartley combine matrices
// ---------------------------------------------------------------------------
__global__ __launch_bounds__(256) void init_mats_k(float* __restrict__ mats) {
  float* C64 = mats;            // 4096
  float* S64 = mats + 4096;     // 4096
  float* NS64 = mats + 8192;    // 4096
  float* Ci  = mats + 12288;    // 512
  float* Si  = mats + 12800;    // 512
  float* NSi = mats + 13312;    // 512
  float* Tr  = mats + 13824;    // 512
  float* Ti  = mats + 14336;    // 512
  const float w = PI2 / 64.f;
  int stride = blockDim.x * gridDim.x;
  for (int idx = blockIdx.x * blockDim.x + threadIdx.x; idx < 4096; idx += stride) {
    int k = idx >> 6, n = idx & 63;
    int m = (k * n) & 63;
    float c = cosf(w * (float)m), s = sinf(w * (float)m);
    C64[idx] = c; S64[idx] = s; NS64[idx] = -s;
  }
  for (int idx = blockIdx.x * blockDim.x + threadIdx.x; idx < 512; idx += stride) {
    int k3 = idx >> 3, mm = idx & 7;
    int m = (k3 * mm) & 63;
    float c = cosf(w * (float)m), s = sinf(w * (float)m);
    Ci[idx] = c; Si[idx] = s; NSi[idx] = -s;
    int k3p = (64 - k3) & 63;
    float sth = sinf(w * (float)k3), cth = cosf(w * (float)k3);
    int mp = (k3p * mm) & 63;
    float cpp_ = cosf(w * (float)mp), spp = sinf(w * (float)mp);
    // r2[k3] = ReF2[k3] - sin(th)*ReF2[k3'] - cos(th)*ImF2[k3']
    Tr[idx] = c - sth * cpp_ + cth * spp;   // coefficient of G2re[m]
    Ti[idx] = s - sth * spp - cth * cpp_;   // coefficient of G2im[m]
  }
}

// ---------------------------------------------------------------------------
// Generic fused two-term transform pass using V_WMMA_F32_16X16X4_F32:
//   X viewed as [outer][Kdim][inner], Y as [outer][64][inner]
//   Y[o,k,i] = sum_n M1[k,n]*X1[o,n,i]  (+ sum_n M2[k,n]*X2[o,n,i] if M2)
// One wave computes a full 64(k) x 16(col) strip: each B fragment is loaded
// from global once per K-step and reused by 4 WMMAs (A fragments from LDS),
// cutting global B traffic 4x vs one-tile-per-wave.
// ---------------------------------------------------------------------------
__global__ __launch_bounds__(256) void dft_pass_k(
    const float* __restrict__ M1, const float* __restrict__ X1,
    const float* __restrict__ M2, const float* __restrict__ X2,
    float* __restrict__ Y, int outer, int Kdim, int inner)
{
  __shared__ float lm1[64 * 64];
  __shared__ float lm2[64 * 64];
  const int tid = threadIdx.x;
  const int msz = 64 * Kdim;
  for (int i = tid; i < msz; i += 256) lm1[i] = M1[i];
  if (M2) { for (int i = tid; i < msz; i += 256) lm2[i] = M2[i]; }
  __syncthreads();

  const int lane = tid & 31;
  const int wave = tid >> 5;
  const int laneLo = lane & 15;
  const int laneHi = lane >> 4;

  const int nCols = outer * inner;            // always a multiple of 16 here
  const int nColTiles = nCols >> 4;
  const int colTile = blockIdx.x * 8 + wave;
  if (colTile >= nColTiles) return;           // uniform per wave (EXEC all-1s)

  // Per-lane column base (columns flatten (o,i); works for any `inner`)
  const int c = colTile * 16 + laneLo;
  const int o = c / inner;
  const int i = c - o * inner;
  const int colBaseX = o * Kdim * inner + i;
  const int colBaseY = o * 64 * inner + i;

  v8f acc[4];
  acc[0] = (v8f){}; acc[1] = (v8f){}; acc[2] = (v8f){}; acc[3] = (v8f){};

  for (int nb = 0; nb < Kdim; nb += 4) {
    const int n0 = nb + 2 * laneHi;
    v2f b;
    b.x = X1[colBaseX + n0 * inner];
    b.y = X1[colBaseX + (n0 + 1) * inner];
#pragma unroll
    for (int kt = 0; kt < 4; ++kt) {
      v2f a;
      a.x = lm1[(kt * 16 + laneLo) * Kdim + n0];
      a.y = lm1[(kt * 16 + laneLo) * Kdim + n0 + 1];
      acc[kt] = __builtin_amdgcn_wmma_f32_16x16x4_f32(false, a, false, b,
                                                      (short)0, acc[kt],
                                                      false, false);
    }
  }
  if (M2) {
    for (int nb = 0; nb < Kdim; nb += 4) {
      const int n0 = nb + 2 * laneHi;
      v2f b;
      b.x = X2[colBaseX + n0 * inner];
      b.y = X2[colBaseX + (n0 + 1) * inner];
#pragma unroll
      for (int kt = 0; kt < 4; ++kt) {
        v2f a;
        a.x = lm2[(kt * 16 + laneLo) * Kdim + n0];
        a.y = lm2[(kt * 16 + laneLo) * Kdim + n0 + 1];
        acc[kt] = __builtin_amdgcn_wmma_f32_16x16x4_f32(false, a, false, b,
                                                        (short)0, acc[kt],
                                                        false, false);
      }
    }
  }
  // D layout: VGPR r -> M = kt*16 + r + 8*laneHi, N = laneLo
#pragma unroll
  for (int kt = 0; kt < 4; ++kt) {
#pragma unroll
    for (int r = 0; r < 8; ++r) {
      const int k = kt * 16 + r + 8 * laneHi;
      Y[colBaseY + k * inner] = acc[kt][r];
    }
  }
}

// ---------------------------------------------------------------------------
// Forward combine: r[k] = Fre[k] - sin(th3)*Fre[sk] - cos(th3)*Fim[sk],
// sk = (k1,k2,(64-k3)&63). Accumulates per-slab sum(r^2) (fixed-order tree),
// writes truncated 8x8x8 corner (unnormalized).
// ---------------------------------------------------------------------------
__global__ __launch_bounds__(256) void fwd_combine_k(
    const float* __restrict__ Fre, const float* __restrict__ Fim,
    float* __restrict__ r8, float* __restrict__ normF, int slab0)
{
  __shared__ float red[256];
  __shared__ float sT[64], cT[64];
  if (threadIdx.x < 64) {
    float a = (PI2 / 64.f) * (float)threadIdx.x;
    sT[threadIdx.x] = sinf(a);
    cT[threadIdx.x] = cosf(a);
  }
  __syncthreads();
  const int s = blockIdx.x;
  const int g = slab0 + s;
  const float* fre = Fre + (size_t)s * 262144;
  const float* fim = Fim + (size_t)s * 262144;
  float sum = 0.f;
  for (int e = threadIdx.x; e < 262144; e += 256) {
    int k3 = e & 63, k2 = (e >> 6) & 63, k1 = e >> 12;
    int k3n = (64 - k3) & 63;
    int en = (k1 << 12) + (k2 << 6) + k3n;
    float r = fre[e] - sT[k3] * fre[en] - cT[k3] * fim[en];
    sum += r * r;
    if ((k1 < 8) & (k2 < 8) & (k3 < 8))
      r8[(size_t)g * 512 + k1 * 64 + k2 * 8 + k3] = r;
  }
  red[threadIdx.x] = sum;
  __syncthreads();
  for (int st = 128; st > 0; st >>= 1) {
    if (threadIdx.x < st) red[threadIdx.x] += red[threadIdx.x + st];
    __syncthreads();
  }
  if (threadIdx.x == 0) normF[g] = sqrtf(red[0]);
}

// ---------------------------------------------------------------------------
// Per-mode channel mix: conv[b,o,k] = 0.5*sum_i[ Xk*(Wk+Wkn) + Xkn*(Wk-Wkn) ]
// with X = r8/normF (zero-guarded), kn = (-k) mod 8 per axis.
// ---------------------------------------------------------------------------
__global__ __launch_bounds__(256) void mode_mix_k(
    const float* __restrict__ r8, const float* __restrict__ normF,
    const float* __restrict__ w1, float* __restrict__ conv)
{
  int t = blockIdx.x * blockDim.x + threadIdx.x;   // 65536 total
  int k = t & 511;
  int bo = t >> 9;
  int o = bo & 31, b = bo >> 5;
  int k1 = (k >> 6) & 7, k2 = (k >> 3) & 7, k3 = k & 7;
  int kn = (((8 - k1) & 7) << 6) | (((8 - k2) & 7) << 3) | ((8 - k3) & 7);
  float acc = 0.f;
  for (int i = 0; i < 32; ++i) {
    float nrm = normF[b * 32 + i];
    float inv = (nrm != 0.f) ? 1.f / nrm : 0.f;
    float Xk = r8[((b * 32 + i) << 9) + k] * inv;
    float Xn = r8[((b * 32 + i) << 9) + kn] * inv;
    float Wk = w1[((i * 32 + o) << 9) + k];
    float Wn = w1[((i * 32 + o) << 9) + kn];
    acc += Xk * (Wk + Wn) + Xn * (Wk - Wn);
  }
  conv[t] = 0.5f * acc;
}

// Per-slab L2 norm of the inverse transform (deterministic tree reduce)
__global__ __launch_bounds__(256) void slab_norm_k(
    const float* __restrict__ y, float* __restrict__ norm2)
{
  __shared__ float red[256];
  const float* p = y + (size_t)blockIdx.x * 262144;
  float s = 0.f;
  for (int e = threadIdx.x; e < 262144; e += 256) { float v = p[e]; s += v * v; }
  red[threadIdx.x] = s;
  __syncthreads();
  for (int st = 128; st > 0; st >>= 1) {
    if (threadIdx.x < st) red[threadIdx.x] += red[threadIdx.x + st];
    __syncthreads();
  }
  if (threadIdx.x == 0) norm2[blockIdx.x] = sqrtf(red[0]);
}

// out = (norm!=0 ? r2/norm : 0) / 262144
__global__ __launch_bounds__(256) void final_scale_k(
    float* __restrict__ y, const float* __restrict__ norm2)
{
  size_t stride = (size_t)blockDim.x * gridDim.x;
  for (size_t idx = (size_t)blockIdx.x * blockDim.x + threadIdx.x;
       idx < (size_t)33554432; idx += stride) {
    int g = (int)(idx >> 18);
    float n = norm2[g];
    y[idx] = (n != 0.f) ? y[idx] / (n * 262144.f) : 0.f;
  }
}

// ---------------------------------------------------------------------------
extern "C" void kernel_launch(void* const* d_in, const int* in_sizes, int n_in,
                              void* d_out, int out_size, void* d_ws, size_t ws_size,
                              hipStream_t stream) {
  const float* x  = (const float*)d_in[0];   // (4,32,64,64,64) f32
  const float* w1 = (const float*)d_in[1];   // (32,32,8,8,8)  f32
  float* out = (float*)d_out;                // (4,32,64,64,64) f32
  float* ws = (float*)d_ws;

  const int CH = 8;                          // slabs per forward chunk (128 total)
  const size_t CHW = (size_t)CH * 262144;

  // Workspace layout (floats), ~72 MB total
  float* mats  = ws;                  // 16384 reserved for tables
  float* A     = ws + 16384;          // Fre ping
  float* Bq    = A + CHW;             // Fim ping
  float* Cq    = Bq + CHW;            // Fre pong
  float* Dq    = Cq + CHW;            // Fim pong
  float* normF = Dq + CHW;            // 128
  float* r8    = normF + 128;         // 4*32*512
  float* convb = r8 + 65536;          // 4*32*512
  float* G1re  = convb + 65536;       // 128*64*8*8
  float* G1im  = G1re + 524288;
  float* G2re  = G1im + 524288;       // 128*64*64*8
  float* G2im  = G2re + 4194304;
  float* norm2 = G2im + 4194304;      // 128

  const float* C64 = mats, *S64 = mats + 4096, *NS64 = mats + 8192;
  const float* Ci = mats + 12288, *Si = mats + 12800, *NSi = mats + 13312;
  const float* Trm = mats + 13824, *Tim = mats + 14336;

  init_mats_k<<<8, 256, 0, stream>>>(ws);

  auto dftGrid = [](int outer, int inner) -> dim3 {
    long long colTiles = (long long)outer * inner / 16;      // one strip per wave
    return dim3((unsigned)((colTiles + 7) / 8));             // 8 waves / block
  };

  // -------- forward: full complex FFT per chunk, then combine+norm+truncate
  for (int ch = 0; ch < 16; ++ch) {
    const float* xc = x + (size_t)ch * CHW;
    // axis z (real input): Fre = C*x ; Fim = -S*x
    dft_pass_k<<<dftGrid(CH*4096,1),256,0,stream>>>(C64,  xc, nullptr, nullptr, A,  CH*4096, 64, 1);
    dft_pass_k<<<dftGrid(CH*4096,1),256,0,stream>>>(NS64, xc, nullptr, nullptr, Bq, CH*4096, 64, 1);
    // axis y (complex): re' = C*re + S*im ; im' = C*im - S*re
    dft_pass_k<<<dftGrid(CH*64,64),256,0,stream>>>(C64, A,  S64,  Bq, Cq, CH*64, 64, 64);
    dft_pass_k<<<dftGrid(CH*64,64),256,0,stream>>>(C64, Bq, NS64, A,  Dq, CH*64, 64, 64);
    // axis x (complex)
    dft_pass_k<<<dftGrid(CH,4096),256,0,stream>>>(C64, Cq, S64,  Dq, A,  CH, 64, 4096);
    dft_pass_k<<<dftGrid(CH,4096),256,0,stream>>>(C64, Dq, NS64, Cq, Bq, CH, 64, 4096);
    fwd_combine_k<<<CH,256,0,stream>>>(A, Bq, r8, normF, ch * CH);
  }

  // -------- per-mode channel mixing (tiny)
  mode_mix_k<<<256,256,0,stream>>>(r8, normF, w1, convb);

  // -------- inverse: expand 8^3 corner to 64^3 with fused combine on axis 3
  dft_pass_k<<<dftGrid(128,64),256,0,stream>>>(Ci,  convb, nullptr, nullptr, G1re, 128, 8, 64);
  dft_pass_k<<<dftGrid(128,64),256,0,stream>>>(NSi, convb, nullptr, nullptr, G1im, 128, 8, 64);
  dft_pass_k<<<dftGrid(8192,8),256,0,stream>>>(Ci, G1re, Si,  G1im, G2re, 8192, 8, 8);
  dft_pass_k<<<dftGrid(8192,8),256,0,stream>>>(Ci, G1im, NSi, G1re, G2im, 8192, 8, 8);
  dft_pass_k<<<dftGrid(524288,1),256,0,stream>>>(Trm, G2re, Tim, G2im, out, 524288, 8, 1);

  // -------- normalize + 1/n scale
  slab_norm_k<<<128,256,0,stream>>>(out, norm2);
  final_scale_k<<<4096,256,0,stream>>>(out, norm2);

  (void)in_sizes; (void)n_in; (void)out_size; (void)ws_size;
}